// VisionMambaModule_6949257085533
// MI455X (gfx1250) — compile-verified
//
#include <hip/hip_runtime.h>
#include <hip/hip_bf16.h>
#include <math.h>

// ---------------------------------------------------------------------------
// VisionMamba module for MI455X (gfx1250, wave32).
// Pipeline: pool -> GEMM(+bias) -> LN -> GEMM -> dwconv+silu -> GEMM ->
//           GEMM(+bias+softplus) -> fused scan(+u*D, *silu(z)) -> GEMM ->
//           bilinear upsample.
// GEMMs use V_WMMA_F32_16X16X4_F32 (exact fp32) with MBxNB register blocking:
// each wave owns an (MB*16)x(NB*16) C tile -> fragment reuse + 4 independent
// WMMA accumulator chains for XDL latency hiding.
// ---------------------------------------------------------------------------

typedef float v2f __attribute__((ext_vector_type(2)));
typedef float v8f __attribute__((ext_vector_type(8)));

#define B_SZ    64
#define C_IN    2048
#define H_SZ    24
#define W_SZ    12
#define HID     768
#define INNER   1536
#define NSTATE  16
#define DTRANK  48
#define LTOK    9
#define MTOK    (B_SZ * LTOK)   // 576 rows

__device__ __forceinline__ float silu_f(float x) {
    return x / (1.0f + __expf(-x));
}

// ---------------------------------------------------------------------------
// 1) 8x4 average pool: x[64,2048,24,12] -> tokA[576,2048] (token-major, K
//    contiguous for the following GEMM's A operand). One thread per (b,c):
//    contiguous 1152B streaming read per thread, 9 coalesced stores.
// ---------------------------------------------------------------------------
__global__ __launch_bounds__(256) void pool_kernel(const float* __restrict__ x,
                                                   float* __restrict__ tokA) {
    int idx = blockIdx.x * blockDim.x + threadIdx.x;       // b*2048 + c
    if (idx >= B_SZ * C_IN) return;
    int c = idx % C_IN;
    int b = idx / C_IN;
    const float4* row = (const float4*)(x + (size_t)idx * (H_SZ * W_SZ));
    // Pull the second half of this thread's 1152B region toward the caches
    // while we chew on the first half (gfx1250 global_prefetch_b8).
    __builtin_prefetch((const void*)(row + 36), 0, 1);
    float acc[3][3] = {};
#pragma unroll
    for (int h = 0; h < H_SZ; ++h) {
        int ph = h >> 3;                                   // 8 input rows per pool row
        float4 r0 = row[h * 3 + 0];
        float4 r1 = row[h * 3 + 1];
        float4 r2 = row[h * 3 + 2];
        acc[ph][0] += r0.x + r0.y + r0.z + r0.w;
        acc[ph][1] += r1.x + r1.y + r1.z + r1.w;
        acc[ph][2] += r2.x + r2.y + r2.z + r2.w;
    }
#pragma unroll
    for (int ph = 0; ph < 3; ++ph)
#pragma unroll
        for (int pw = 0; pw < 3; ++pw)
            tokA[((size_t)(b * LTOK + ph * 3 + pw)) * C_IN + c] = acc[ph][pw] * (1.0f / 32.0f);
}

// ---------------------------------------------------------------------------
// 2) Register-blocked fp32 WMMA GEMM: C = A[M,K] @ B[K,N] (+ epilogue).
//    One wave per (MB*16)x(NB*16) C tile; K stepped by 4 through
//    V_WMMA_F32_16X16X4_F32. Requires M%(16*MB)==0, N%(16*NB)==0, K%4==0.
//    Fragment layouts per CDNA5 ISA 7.12.2:
//      A 16x4 : lane l holds A[m0 + l%16][kb + 2*(l/16) .. +1]
//      B 4x16 : lane l holds B[kb + 2*(l/16) .. +1][n0 + l%16]
//      C 16x16: vgpr v of lane l = C[m0 + v + 8*(l/16)][n0 + l%16]
//    EPI: 0 = none, 1 = +bias, 2 = +bias then softplus.
// ---------------------------------------------------------------------------
template <int MB, int NB, int EPI>
__global__ __launch_bounds__(32) void gemm_wmma_f32(const float* __restrict__ A,
                                                    const float* __restrict__ Bm,
                                                    const float* __restrict__ bias,
                                                    float* __restrict__ C,
                                                    int K, int lda, int ldb, int ldc) {
    const int n0   = blockIdx.x * (16 * NB);
    const int m0   = blockIdx.y * (16 * MB);
    const int lane = threadIdx.x & 31;
    const int r    = lane & 15;        // row-in-tile (A) / col-in-tile (B,C)
    const int hi   = lane >> 4;        // K sub-pair selector / M+8 selector

    const float* arow = A + (size_t)(m0 + r) * lda + 2 * hi;
    const float* bcol = Bm + (size_t)(2 * hi) * ldb + n0 + r;

    v8f acc[MB][NB] = {};

    for (int kb = 0; kb < K; kb += 4) {
        v2f a[MB];
        v2f b[NB];
#pragma unroll
        for (int i = 0; i < MB; ++i) {
            const float* ap = arow + (size_t)(i * 16) * lda + kb;
            a[i].x = ap[0];
            a[i].y = ap[1];
        }
#pragma unroll
        for (int j = 0; j < NB; ++j) {
            const float* bp = bcol + (size_t)kb * ldb + j * 16;
            b[j].x = bp[0];
            b[j].y = bp[ldb];
        }
#pragma unroll
        for (int i = 0; i < MB; ++i)
#pragma unroll
            for (int j = 0; j < NB; ++j)
                acc[i][j] = __builtin_amdgcn_wmma_f32_16x16x4_f32(
                    /*neg_a=*/false, a[i], /*neg_b=*/false, b[j],
                    /*c_mod=*/(short)0, acc[i][j],
                    /*reuse_a=*/false, /*reuse_b=*/false);
    }

#pragma unroll
    for (int j = 0; j < NB; ++j) {
        const int n = n0 + j * 16 + r;
        float bv = 0.0f;
        if (EPI >= 1) bv = bias[n];
#pragma unroll
        for (int i = 0; i < MB; ++i) {
#pragma unroll
            for (int v = 0; v < 8; ++v) {
                int m = m0 + i * 16 + v + 8 * hi;
                float val = acc[i][j][v] + bv;
                if (EPI == 2) val = (val > 20.0f) ? val : log1pf(__expf(val));
                C[(size_t)m * ldc + n] = val;
            }
        }
    }
}

// ---------------------------------------------------------------------------
// 3) LayerNorm over HID=768, in place. One 256-thread block per token row.
// ---------------------------------------------------------------------------
__global__ __launch_bounds__(256) void ln_kernel(float* __restrict__ t,
                                                 const float* __restrict__ g,
                                                 const float* __restrict__ b) {
    __shared__ float red[256];
    const int tid = threadIdx.x;
    float* p = t + (size_t)blockIdx.x * HID;

    float v[3];
    float s = 0.0f;
#pragma unroll
    for (int i = 0; i < 3; ++i) { v[i] = p[tid + 256 * i]; s += v[i]; }
    red[tid] = s;
    __syncthreads();
    for (int off = 128; off > 0; off >>= 1) {
        if (tid < off) red[tid] += red[tid + off];
        __syncthreads();
    }
    const float mu = red[0] * (1.0f / HID);
    __syncthreads();

    float s2 = 0.0f;
#pragma unroll
    for (int i = 0; i < 3; ++i) { float d = v[i] - mu; s2 += d * d; }
    red[tid] = s2;
    __syncthreads();
    for (int off = 128; off > 0; off >>= 1) {
        if (tid < off) red[tid] += red[tid + off];
        __syncthreads();
    }
    const float rstd = rsqrtf(red[0] * (1.0f / HID) + 1e-5f);

#pragma unroll
    for (int i = 0; i < 3; ++i) {
        int col = tid + 256 * i;
        p[col] = (v[i] - mu) * rstd * g[col] + b[col];
    }
}

// ---------------------------------------------------------------------------
// 4) Depthwise conv (k=3, pad=1) along L=9 + SiLU.
//    xc = first INNER columns of xz[576,3072]. One thread per (token, d).
// ---------------------------------------------------------------------------
__global__ __launch_bounds__(256) void conv_silu_kernel(const float* __restrict__ xz,
                                                        const float* __restrict__ w,
                                                        const float* __restrict__ cb,
                                                        float* __restrict__ u) {
    int idx = blockIdx.x * blockDim.x + threadIdx.x;       // t*INNER + d
    if (idx >= MTOK * INNER) return;
    int d = idx % INNER;
    int t = idx / INNER;
    int l = t % LTOK;
    int b = t / LTOK;
    const float* base = xz + (size_t)(b * LTOK) * (2 * INNER) + d;
    float acc = cb[d];
    if (l > 0)        acc += base[(size_t)(l - 1) * (2 * INNER)] * w[d * 3 + 0];
                      acc += base[(size_t)l       * (2 * INNER)] * w[d * 3 + 1];
    if (l < LTOK - 1) acc += base[(size_t)(l + 1) * (2 * INNER)] * w[d * 3 + 2];
    u[idx] = silu_f(acc);
}

// ---------------------------------------------------------------------------
// 5) Selective scan over L=9, fused with +u*D and *silu(z).
//    One thread per (b,d): 16 states + 16 A coefficients in registers.
//    B/C per step come from dBC[:,48:64] / dBC[:,64:80] (L2-resident, 180KB).
// ---------------------------------------------------------------------------
__global__ __launch_bounds__(256) void scan_kernel(const float* __restrict__ u,
                                                   const float* __restrict__ delta,
                                                   const float* __restrict__ dBC,
                                                   const float* __restrict__ A_log,
                                                   const float* __restrict__ Dp,
                                                   const float* __restrict__ xz,
                                                   float* __restrict__ y) {
    int idx = blockIdx.x * blockDim.x + threadIdx.x;       // b*INNER + d
    if (idx >= B_SZ * INNER) return;
    int d = idx % INNER;
    int b = idx / INNER;

    float Ar[NSTATE], s[NSTATE];
#pragma unroll
    for (int n = 0; n < NSTATE; ++n) {
        Ar[n] = -__expf(A_log[d * NSTATE + n]);
        s[n]  = 0.0f;
    }
    const float Dd = Dp[d];

    for (int l = 0; l < LTOK; ++l) {
        int t = b * LTOK + l;
        float ul = u[(size_t)t * INNER + d];
        float dl = delta[(size_t)t * INNER + d];
        const float* bc = dBC + (size_t)t * (DTRANK + 2 * NSTATE);
        float yv = 0.0f;
#pragma unroll
        for (int n = 0; n < NSTATE; ++n) {
            float k1 = Ar[n] * s[n] + bc[DTRANK + n] * ul;   // A*s + B*u
            s[n] += dl * k1;                                 // Euler update
            yv  += s[n] * bc[DTRANK + NSTATE + n];           // dot with C
        }
        float zv = xz[(size_t)t * (2 * INNER) + INNER + d];
        y[(size_t)t * INNER + d] = (yv + ul * Dd) * silu_f(zv);
    }
}

// ---------------------------------------------------------------------------
// 6) Bilinear upsample 3x3 -> 24x12. One thread per output element (writes
//    coalesced, 4 gather reads of the L2-resident 4.7MB g buffer).
// ---------------------------------------------------------------------------
__global__ __launch_bounds__(256) void upsample_kernel(const float* __restrict__ g,
                                                       float* __restrict__ out) {
    size_t idx = (size_t)blockIdx.x * blockDim.x + threadIdx.x;
    const size_t total = (size_t)B_SZ * C_IN * H_SZ * W_SZ;
    if (idx >= total) return;
    int w = (int)(idx % W_SZ);
    size_t t1 = idx / W_SZ;
    int h = (int)(t1 % H_SZ);
    size_t t2 = t1 / H_SZ;
    int c = (int)(t2 % C_IN);
    int b = (int)(t2 / C_IN);

    float sy = (h + 0.5f) * (3.0f / H_SZ) - 0.5f;
    sy = fmaxf(sy, 0.0f);
    int i0y = min((int)sy, 2);
    int i1y = min(i0y + 1, 2);
    float wy = sy - (float)i0y;

    float sx = (w + 0.5f) * (3.0f / W_SZ) - 0.5f;
    sx = fmaxf(sx, 0.0f);
    int i0x = min((int)sx, 2);
    int i1x = min(i0x + 1, 2);
    float wx = sx - (float)i0x;

    const size_t bb = (size_t)b * LTOK;
    #define GVAL(gy, gx) g[(bb + (gy) * 3 + (gx)) * C_IN + c]
    float v00 = GVAL(i0y, i0x);
    float v01 = GVAL(i0y, i1x);
    float v10 = GVAL(i1y, i0x);
    float v11 = GVAL(i1y, i1x);
    #undef GVAL

    out[idx] = (1.0f - wy) * ((1.0f - wx) * v00 + wx * v01) +
               wy * ((1.0f - wx) * v10 + wx * v11);
}

// ---------------------------------------------------------------------------
// Launch
// ---------------------------------------------------------------------------
extern "C" void kernel_launch(void* const* d_in, const int* in_sizes, int n_in,
                              void* d_out, int out_size, void* d_ws, size_t ws_size,
                              hipStream_t stream) {
    const float* x      = (const float*)d_in[0];
    const float* pe_W   = (const float*)d_in[1];
    const float* pe_b   = (const float*)d_in[2];
    const float* ln_g   = (const float*)d_in[3];
    const float* ln_b   = (const float*)d_in[4];
    const float* in_W   = (const float*)d_in[5];
    const float* conv_w = (const float*)d_in[6];
    const float* conv_b = (const float*)d_in[7];
    const float* xprojW = (const float*)d_in[8];
    const float* dt_W   = (const float*)d_in[9];
    const float* dt_b   = (const float*)d_in[10];
    const float* A_log  = (const float*)d_in[11];
    const float* D_par  = (const float*)d_in[12];
    const float* out_W  = (const float*)d_in[13];
    float* out = (float*)d_out;

    // Workspace carve-up (floats). Total ~27.8 MB (fully L2-resident).
    float* ws    = (float*)d_ws;
    float* tokA  = ws;                              // [576, 2048]
    float* tok   = tokA  + (size_t)MTOK * C_IN;     // [576, 768]
    float* xz    = tok   + (size_t)MTOK * HID;      // [576, 3072]
    float* u     = xz    + (size_t)MTOK * 2 * INNER;// [576, 1536]
    float* dBC   = u     + (size_t)MTOK * INNER;    // [576, 80]
    float* delta = dBC   + (size_t)MTOK * (DTRANK + 2 * NSTATE); // [576, 1536]
    float* ybuf  = delta + (size_t)MTOK * INNER;    // [576, 1536]
    float* gbuf  = ybuf  + (size_t)MTOK * INNER;    // [576, 2048]

    // 1) pool
    pool_kernel<<<(B_SZ * C_IN + 255) / 256, 256, 0, stream>>>(x, tokA);

    // 2) tok = tokA @ pe_W + pe_b   [576,2048]x[2048,768], 32x32 tiles
    gemm_wmma_f32<2, 2, 1><<<dim3(HID / 32, MTOK / 32), 32, 0, stream>>>(
        tokA, pe_W, pe_b, tok, C_IN, C_IN, HID, HID);

    // 3) LayerNorm in place
    ln_kernel<<<MTOK, 256, 0, stream>>>(tok, ln_g, ln_b);

    // 4) xz = tok @ in_W            [576,768]x[768,3072], 32x32 tiles
    gemm_wmma_f32<2, 2, 0><<<dim3((2 * INNER) / 32, MTOK / 32), 32, 0, stream>>>(
        tok, in_W, nullptr, xz, HID, HID, 2 * INNER, 2 * INNER);

    // 5) depthwise conv + silu -> u
    conv_silu_kernel<<<(MTOK * INNER + 255) / 256, 256, 0, stream>>>(xz, conv_w, conv_b, u);

    // 6) dBC = u @ xproj_W          [576,1536]x[1536,80], 32x16 tiles (N=80)
    gemm_wmma_f32<2, 1, 0><<<dim3((DTRANK + 2 * NSTATE) / 16, MTOK / 32), 32, 0, stream>>>(
        u, xprojW, nullptr, dBC, INNER,
        INNER, DTRANK + 2 * NSTATE, DTRANK + 2 * NSTATE);

    // 7) delta = softplus(dBC[:, :48] @ dt_W + dt_b)   [576,48]x[48,1536]
    gemm_wmma_f32<2, 2, 2><<<dim3(INNER / 32, MTOK / 32), 32, 0, stream>>>(
        dBC, dt_W, dt_b, delta, DTRANK,
        DTRANK + 2 * NSTATE, INNER, INNER);

    // 8) fused selective scan (+u*D, *silu(z)) -> ybuf
    scan_kernel<<<(B_SZ * INNER + 255) / 256, 256, 0, stream>>>(
        u, delta, dBC, A_log, D_par, xz, ybuf);

    // 9) g = ybuf @ out_W           [576,1536]x[1536,2048], 32x32 tiles
    gemm_wmma_f32<2, 2, 0><<<dim3(C_IN / 32, MTOK / 32), 32, 0, stream>>>(
        ybuf, out_W, nullptr, gbuf, INNER, INNER, C_IN, C_IN);

    // 10) bilinear upsample 3x3 -> 24x12
    const size_t total = (size_t)B_SZ * C_IN * H_SZ * W_SZ;
    upsample_kernel<<<(unsigned)((total + 255) / 256), 256, 0, stream>>>(gbuf, out);
}